// TrajectoryCritic_47236050322110
// MI455X (gfx1250) — compile-verified
//
#include <hip/hip_runtime.h>
#include <hip/hip_bf16.h>

// Problem constants (match reference)
#define S_LEN   20
#define NGRP    64
#define NPED    128
#define NTOT    (NGRP * NPED)     // 8192
#define MPTS    200
#define KPOS    (S_LEN * NPED)    // 2560
#define MLP_DIM 1024
#define COLL_THR2 0.0625f         // 0.25^2
#define OCC_THR2  1.0f            // 1.0^2

typedef __attribute__((ext_vector_type(2))) float v2f;
typedef __attribute__((ext_vector_type(8))) float v8f;

// ---------------------------------------------------------------------------
// Kernel 1: per-agent collision counts via V_WMMA_F32_16X16X4_F32.
// One block per group (256 thr = 8 waves). Wave w owns row-tile [w*16, w*16+16).
// d^2(i,j) = A x B with A row i = (x_i, y_i, |x_i|^2, 1),
//                      B col j = (-2x_j, -2y_j, 1, |x_j|^2).
// ---------------------------------------------------------------------------
__global__ __launch_bounds__(256) void agent_count_kernel(
    const float* __restrict__ traj, float* __restrict__ cnt_a) {
  const int g = blockIdx.x;
  __shared__ float lx[NPED], ly[NPED], ls[NPED];

  const int tid  = threadIdx.x;
  const int wave = tid >> 5;
  const int lane = tid & 31;
  const int half = lane >> 4;   // 0: lanes 0-15, 1: lanes 16-31
  const int l16  = lane & 15;
  const int i0   = wave * 16;   // row-tile base (pedestrian index in group)

  float acc[8];
#pragma unroll
  for (int v = 0; v < 8; ++v) acc[v] = 0.0f;

  for (int t = 0; t < S_LEN; ++t) {
    if (tid < NPED) {
      const int idx = (t * NTOT + g * NPED + tid) * 2;
      const float x = traj[idx];
      const float y = traj[idx + 1];
      lx[tid] = x; ly[tid] = y; ls[tid] = x * x + y * y;
    }
    __syncthreads();

    // A operand: 16x4 f32, lanes 0-15 -> K=0,1 ; lanes 16-31 -> K=2,3
    const int ia = i0 + l16;
    v2f a;
    a[0] = half ? ls[ia] : lx[ia];
    a[1] = half ? 1.0f   : ly[ia];

#pragma unroll
    for (int jt = 0; jt < 8; ++jt) {
      const int jb = jt * 16 + l16;
      // B operand: 4x16 f32, lanes 0-15 -> K=0,1 rows ; lanes 16-31 -> K=2,3
      v2f b;
      b[0] = half ? 1.0f   : -2.0f * lx[jb];
      b[1] = half ? ls[jb] : -2.0f * ly[jb];

      v8f c = {};
      // D = A x B + 0 : 16x16 tile of squared distances
      v8f d = __builtin_amdgcn_wmma_f32_16x16x4_f32(
          /*neg_a=*/false, a, /*neg_b=*/false, b,
          /*c_mod=*/(short)0, c, /*reuse_a=*/false, /*reuse_b=*/false);

#pragma unroll
      for (int v = 0; v < 8; ++v) {
        const int m = v + half * 8;    // row within tile
        const int i = i0 + m;          // pedestrian i
        const int j = jt * 16 + l16;   // pedestrian j (column)
        const float d2 = d[v];
        if (d2 > 0.0f && d2 < COLL_THR2 && i != j) acc[v] += 1.0f;
      }
    }
    __syncthreads();
  }

  // Reduce over the 16 columns: xor-tree within each 16-lane half.
#pragma unroll
  for (int v = 0; v < 8; ++v) {
    float r = acc[v];
    r += __shfl_xor(r, 8, 32);
    r += __shfl_xor(r, 4, 32);
    r += __shfl_xor(r, 2, 32);
    r += __shfl_xor(r, 1, 32);
    if (l16 == 0) {
      const int i = i0 + v + half * 8;
      cnt_a[g * NPED + i] = r;
    }
  }
}

// ---------------------------------------------------------------------------
// Kernel 2: scene (occupancy) counts, faithful to the repeat/view indexing:
//   F in [0, KPOS*MPTS); si = F / KPOS; pi = F / MPTS; ped = F % 128.
// Thread t walks F = t, t+256, ... so ped == t & 127 is constant per thread.
// pos & scene staged in LDS.
// ---------------------------------------------------------------------------
__global__ __launch_bounds__(256) void scene_count_kernel(
    const float* __restrict__ traj, const int* __restrict__ seq_scene_ids,
    const float* __restrict__ scenes, float* __restrict__ cnt_s) {
  const int g = blockIdx.x;
  __shared__ float px[KPOS], py[KPOS];
  __shared__ float sx[MPTS], sy[MPTS];
  __shared__ float cbin[256];

  const int tid = threadIdx.x;
  const int sid = seq_scene_ids[g];

  for (int i = tid; i < MPTS; i += 256) {
    sx[i] = scenes[(sid * MPTS + i) * 2];
    sy[i] = scenes[(sid * MPTS + i) * 2 + 1];
  }
  for (int k = tid; k < KPOS; k += 256) {
    const int time = k >> 7;       // k / 128
    const int p    = k & 127;      // k % 128
    const int idx  = (time * NTOT + g * NPED + p) * 2;
    px[k] = traj[idx];
    py[k] = traj[idx + 1];
  }
  __syncthreads();

  float cnt = 0.0f;
  const unsigned FMAX = (unsigned)(KPOS * MPTS);   // 512000
  for (unsigned F = (unsigned)tid; F < FMAX; F += 256u) {
    const unsigned si = F / (unsigned)KPOS;   // scene point actually read
    const unsigned pi = F / (unsigned)MPTS;   // flattened (t,ped) position
    const float dx = sx[si] - px[pi];
    const float dy = sy[si] - py[pi];
    const float d2 = dx * dx + dy * dy;
    cnt += (d2 < OCC_THR2) ? 1.0f : 0.0f;
  }
  cbin[tid] = cnt;
  __syncthreads();
  if (tid < NPED) cnt_s[g * NPED + tid] = cbin[tid] + cbin[tid + NPED];
}

// ---------------------------------------------------------------------------
// Kernel 3: scalar-input MLP  score = relu(c*w1 + b1) @ w2 + b2,
// thread-per-pedestrian, weights broadcast from LDS.
// ---------------------------------------------------------------------------
__global__ __launch_bounds__(256) void mlp_kernel(
    const float* __restrict__ cnt, const float* __restrict__ w1,
    const float* __restrict__ b1, const float* __restrict__ w2,
    const float* __restrict__ b2, float* __restrict__ out) {
  __shared__ float sw1[MLP_DIM], sb1[MLP_DIM], sw2[MLP_DIM];
  const int tid = threadIdx.x;
  for (int k = tid; k < MLP_DIM; k += 256) {
    sw1[k] = w1[k];
    sb1[k] = b1[k];
    sw2[k] = w2[k];
  }
  __syncthreads();

  const int ped = blockIdx.x * 256 + tid;
  const float c = cnt[ped];
  float acc = 0.0f;
#pragma unroll 4
  for (int k = 0; k < MLP_DIM; ++k) {
    const float h = fmaf(c, sw1[k], sb1[k]);
    acc = fmaf(fmaxf(h, 0.0f), sw2[k], acc);
  }
  out[ped] = acc + b2[0];
}

// ---------------------------------------------------------------------------
// Launch: inputs in setup_inputs() order.
//   0 traj, 1 traj_rel, 2 seq_start_end, 3 seq_scene_ids, 4 scenes,
//   5 w1a, 6 b1a, 7 w2a, 8 b2a, 9 w1s, 10 b1s, 11 w2s, 12 b2s
// d_out = [scores1 (8192) | scores2 (8192)] f32.
// ---------------------------------------------------------------------------
extern "C" void kernel_launch(void* const* d_in, const int* in_sizes, int n_in,
                              void* d_out, int out_size, void* d_ws, size_t ws_size,
                              hipStream_t stream) {
  const float* traj          = (const float*)d_in[0];
  const int*   seq_scene_ids = (const int*)d_in[3];
  const float* scenes        = (const float*)d_in[4];
  const float* w1a = (const float*)d_in[5];
  const float* b1a = (const float*)d_in[6];
  const float* w2a = (const float*)d_in[7];
  const float* b2a = (const float*)d_in[8];
  const float* w1s = (const float*)d_in[9];
  const float* b1s = (const float*)d_in[10];
  const float* w2s = (const float*)d_in[11];
  const float* b2s = (const float*)d_in[12];

  float* out   = (float*)d_out;
  float* cnt_a = (float*)d_ws;          // NTOT floats
  float* cnt_s = cnt_a + NTOT;          // NTOT floats

  agent_count_kernel<<<NGRP, 256, 0, stream>>>(traj, cnt_a);
  scene_count_kernel<<<NGRP, 256, 0, stream>>>(traj, seq_scene_ids, scenes, cnt_s);
  mlp_kernel<<<NTOT / 256, 256, 0, stream>>>(cnt_a, w1a, b1a, w2a, b2a, out);
  mlp_kernel<<<NTOT / 256, 256, 0, stream>>>(cnt_s, w1s, b1s, w2s, b2s, out + NTOT);
}